// MLP_32083405701774
// MI455X (gfx1250) — compile-verified
//
#include <hip/hip_runtime.h>

typedef __attribute__((ext_vector_type(2))) float v2f;
typedef __attribute__((ext_vector_type(4))) float v4f;
typedef __attribute__((ext_vector_type(8))) float v8f;

#define NSAMP 100000   // N
#define INDIM 500      // IN
#define HDIM  256      // H
#define MROWS 50000    // M
#define KCLUS 50000    // K (clusters)
#define OUTSQ 1024     // OUT*OUT
#define CATK  512      // 2*H

// ---------------------------------------------------------------------------
// Generic WMMA f32 GEMM: Z[rows, Ntot] = op(A)[rows, K] @ W[Ntot, K]^T + bias
// MODE 0: A read directly                         (GEMM1: x @ W1^T)
// MODE 1: A = relu(Araw*scale[k] + shift[k])      (GEMM2: bn-relu(z1) @ W2^T)
// MODE 2: A = cat(xc, x1) with half-swap gather   (GEMM3: cat @ Wfc^T)
// STATS : accumulate per-column sum / sum-of-squares of Z (for BatchNorm)
// One wave computes a 16-row x 256-col strip (16 WMMA accum tiles).
// Block = 4 waves = 64 rows. blockIdx.y selects a 256-col chunk of Ntot.
// ---------------------------------------------------------------------------
template<int MODE, bool STATS>
__global__ __launch_bounds__(128) void wmma_gemm(
    const float* __restrict__ A,  const float* __restrict__ A2,
    const float* __restrict__ W,  const float* __restrict__ bias,
    const float* __restrict__ scale, const float* __restrict__ shift,
    float* __restrict__ Z, float* __restrict__ gsum, float* __restrict__ gsq,
    int rows, int K, int Ntot)
{
    __shared__ float lsum[256];
    __shared__ float lsq[256];
    __shared__ __align__(16) float ltile[4 * 16 * 128];   // 32 KB

    const int tid  = threadIdx.x;
    const int wave = tid >> 5;
    const int lane = tid & 31;
    const int lm   = lane & 15;     // column-within-tile / row-within-tile
    const int lh   = lane >> 4;     // half-wave selects K pair (f32 wmma A/B layout)
    const int nbase = blockIdx.y * 256;
    const int r0    = (blockIdx.x * 4 + wave) * 16;
    const bool active = (r0 < rows);

    if (STATS) {
        for (int i = tid; i < 256; i += 128) { lsum[i] = 0.f; lsq[i] = 0.f; }
        __syncthreads();
    }

    v8f acc[16];
    if (active) {
        const v8f vzero = {0.f,0.f,0.f,0.f,0.f,0.f,0.f,0.f};
#pragma unroll
        for (int nt = 0; nt < 16; ++nt) acc[nt] = vzero;

        // Per-lane A addressing (row r0+lm; K-offset pair chosen by lh)
        const float* aRow    = nullptr;
        const float* aFirst  = nullptr;   // MODE 2: source for cols [0,256)
        const float* aSecond = nullptr;   // MODE 2: source for cols [256,512)
        if (MODE == 2) {
            const int half = rows >> 1;          // == M
            const int rr   = r0 + lm;
            const int m    = (rr < half) ? rr : rr - half;
            const bool swp = (rr >= half);
            aFirst  = (swp ? A2 : A) + (size_t)m * HDIM;
            aSecond = (swp ? A  : A2) + (size_t)m * HDIM;
        } else {
            aRow = A + (size_t)(r0 + lm) * K;
        }

        for (int k = 0; k < K; k += 4) {
            const int kc = k + lh * 2;
            v2f a;
            if (MODE == 2) {
                const float* p = (kc < 256) ? aFirst : aSecond;
                a = *(const v2f*)(p + (kc & 255));
            } else {
                a = *(const v2f*)(aRow + kc);
                if (MODE == 1) {
                    v2f sc = *(const v2f*)(scale + kc);
                    v2f sh = *(const v2f*)(shift + kc);
                    a.x = fmaxf(a.x * sc.x + sh.x, 0.f);
                    a.y = fmaxf(a.y * sc.y + sh.y, 0.f);
                }
            }
#pragma unroll
            for (int nt = 0; nt < 16; ++nt) {
                v2f b = *(const v2f*)(W + (size_t)(nbase + nt * 16 + lm) * K + kc);
                acc[nt] = __builtin_amdgcn_wmma_f32_16x16x4_f32(
                    false, a, false, b, (short)0, acc[nt], false, false);
            }
        }

        // Bias + per-column BN statistics (each lane holds one fixed column per tile)
#pragma unroll
        for (int nt = 0; nt < 16; ++nt) {
            const int col = nt * 16 + lm;
            const float bv = bias[nbase + col];
            float s = 0.f, q = 0.f;
#pragma unroll
            for (int r = 0; r < 8; ++r) {
                float v = acc[nt][r] + bv;
                acc[nt][r] = v;
                s += v; q += v * v;
            }
            if (STATS) {
                atomicAdd(&lsum[col], s);   // ds_add_f32
                atomicAdd(&lsq[col],  q);
            }
        }
    }

    if (STATS) {
        __syncthreads();
        for (int i = tid; i < 256; i += 128) {
            atomicAdd(&gsum[i], lsum[i]);
            atomicAdd(&gsq[i],  lsq[i]);
        }
    }

    if (active) {
        // Transpose the wave's 16x256 strip through LDS for coalesced b128 stores
        float* my = ltile + wave * (16 * 128);
#pragma unroll
        for (int half = 0; half < 2; ++half) {
#pragma unroll
            for (int t = 0; t < 8; ++t) {
                const int nt = half * 8 + t;
#pragma unroll
                for (int r = 0; r < 8; ++r)
                    my[(r + 8 * lh) * 128 + t * 16 + lm] = acc[nt][r];
            }
            asm volatile("s_wait_dscnt 0x0" ::: "memory");  // LDS writes visible
#pragma unroll
            for (int it = 0; it < 16; ++it) {
                const int i    = (it * 32 + lane) * 4;   // [0, 2048)
                const int row  = i >> 7;
                const int colq = i & 127;
                v4f val = *(const v4f*)(my + i);
                *(v4f*)(Z + (size_t)(r0 + row) * Ntot + nbase + half * 128 + colq) = val;
            }
        }
    }
}

// --- helpers ----------------------------------------------------------------

__global__ void zero_f32(float* __restrict__ p, size_t n) {
    size_t i = (size_t)blockIdx.x * blockDim.x + threadIdx.x;
    const size_t stride = (size_t)gridDim.x * blockDim.x;
    for (; i < n; i += stride) p[i] = 0.f;
}

// One streaming (nontemporal) pass over the 10 GB one-hot matrix -> labels
__global__ void scan_onehot(const float* __restrict__ cid, int* __restrict__ labels) {
    const size_t nvec = (size_t)MROWS * KCLUS / 4;   // K % 4 == 0: vec never crosses rows
    const size_t stride = (size_t)gridDim.x * blockDim.x;
    for (size_t i = (size_t)blockIdx.x * blockDim.x + threadIdx.x; i < nvec; i += stride) {
        v4f v = __builtin_nontemporal_load((const v4f*)cid + i);
        if (v.x != 0.f || v.y != 0.f || v.z != 0.f || v.w != 0.f) {
            const size_t base = i * 4;
            const int m  = (int)(base / KCLUS);
            const int k0 = (int)(base % KCLUS);
            if (v.x != 0.f) labels[m] = k0;
            if (v.y != 0.f) labels[m] = k0 + 1;
            if (v.z != 0.f) labels[m] = k0 + 2;
            if (v.w != 0.f) labels[m] = k0 + 3;
        }
    }
}

__global__ void count_labels(const int* __restrict__ labels, float* __restrict__ counts) {
    const int m = blockIdx.x * blockDim.x + threadIdx.x;
    if (m < MROWS) atomicAdd(&counts[labels[m]], 1.0f);
}

__global__ void finalize_stats(const float* __restrict__ sum, const float* __restrict__ sq,
                               const float* __restrict__ g, const float* __restrict__ beta,
                               float* __restrict__ scale, float* __restrict__ shift, float invN) {
    const int h = threadIdx.x;            // 256 threads
    const float m  = sum[h] * invN;
    const float v  = sq[h] * invN - m * m;     // biased variance (jnp.var)
    const float sc = g[h] * rsqrtf(v + 1e-5f);
    scale[h] = sc;
    shift[h] = beta[h] - m * sc;
}

// xc[m] = relu(bn2(z2[cluster_index[m]]));  segment-sum into csum[labels[m]]
__global__ void gather_csum(const float* __restrict__ z2, const int* __restrict__ cidx,
                            const int* __restrict__ labels, const float* __restrict__ scale,
                            const float* __restrict__ shift, float* __restrict__ xc,
                            float* __restrict__ csum) {
    const int m = blockIdx.x, h = threadIdx.x;
    const int src = cidx[m];
    float v = z2[(size_t)src * HDIM + h];
    v = fmaxf(v * scale[h] + shift[h], 0.f);
    xc[(size_t)m * HDIM + h] = v;
    atomicAdd(&csum[(size_t)labels[m] * HDIM + h], v);
}

// x1[m] = cluster_features[labels[m]] = csum[labels[m]] / counts[labels[m]]
__global__ void make_x1(const float* __restrict__ csum, const float* __restrict__ counts,
                        const int* __restrict__ labels, float* __restrict__ x1) {
    const int m = blockIdx.x, h = threadIdx.x;
    const int l = labels[m];
    x1[(size_t)m * HDIM + h] = csum[(size_t)l * HDIM + h] / counts[l];
}

// ---------------------------------------------------------------------------

extern "C" void kernel_launch(void* const* d_in, const int* in_sizes, int n_in,
                              void* d_out, int out_size, void* d_ws, size_t ws_size,
                              hipStream_t stream) {
    const float* x    = (const float*)d_in[0];
    const float* cid  = (const float*)d_in[1];
    const int*   cidx = (const int*)  d_in[2];
    const float* W1   = (const float*)d_in[3];
    const float* b1   = (const float*)d_in[4];
    const float* g1   = (const float*)d_in[5];
    const float* be1  = (const float*)d_in[6];
    const float* W2   = (const float*)d_in[7];
    const float* b2   = (const float*)d_in[8];
    const float* g2   = (const float*)d_in[9];
    const float* be2  = (const float*)d_in[10];
    const float* Wfc  = (const float*)d_in[11];
    const float* bfc  = (const float*)d_in[12];
    float* out = (float*)d_out;

    // workspace layout (floats)
    float* ws = (float*)d_ws;
    size_t o = 0;
    float* z1     = ws + o; o += (size_t)NSAMP * HDIM;   // 25.6M
    float* z2     = ws + o; o += (size_t)NSAMP * HDIM;   // 25.6M
    float* xc     = ws + o; o += (size_t)MROWS * HDIM;   // 12.8M
    float* x1     = ws + o; o += (size_t)MROWS * HDIM;   // 12.8M
    float* csum   = ws + o; o += (size_t)KCLUS * HDIM;   // 12.8M
    float* counts = ws + o; o += KCLUS;
    float* sum1   = ws + o; o += HDIM;
    float* sq1    = ws + o; o += HDIM;
    float* scale1 = ws + o; o += HDIM;
    float* shift1 = ws + o; o += HDIM;
    float* sum2   = ws + o; o += HDIM;
    float* sq2    = ws + o; o += HDIM;
    float* scale2 = ws + o; o += HDIM;
    float* shift2 = ws + o; o += HDIM;
    int*   labels = (int*)(ws + o); o += MROWS;

    // zero all atomic accumulators (csum, counts, BN stat sums are contiguous)
    const size_t zlen = (size_t)KCLUS * HDIM + KCLUS + 8 * HDIM;
    zero_f32<<<2048, 256, 0, stream>>>(csum, zlen);

    // recover segment structure from the one-hot matrix (single streaming pass)
    scan_onehot<<<4096, 256, 0, stream>>>(cid, labels);
    count_labels<<<(MROWS + 255) / 256, 256, 0, stream>>>(labels, counts);

    const int gx = (NSAMP / 16 + 3) / 4;   // 1563 blocks of 4 waves x 16 rows

    // GEMM1: z1 = x @ W1^T + b1, accumulate BN1 stats
    wmma_gemm<0, true><<<dim3(gx, 1), 128, 0, stream>>>(
        x, nullptr, W1, b1, nullptr, nullptr, z1, sum1, sq1, NSAMP, INDIM, HDIM);
    finalize_stats<<<1, 256, 0, stream>>>(sum1, sq1, g1, be1, scale1, shift1, 1.f / NSAMP);

    // GEMM2: z2 = relu(bn1(z1)) @ W2^T + b2, accumulate BN2 stats
    wmma_gemm<1, true><<<dim3(gx, 1), 128, 0, stream>>>(
        z1, nullptr, W2, b2, scale1, shift1, z2, sum2, sq2, NSAMP, HDIM, HDIM);
    finalize_stats<<<1, 256, 0, stream>>>(sum2, sq2, g2, be2, scale2, shift2, 1.f / NSAMP);

    // xc gather + segment sums, then cluster means gathered back to x1
    gather_csum<<<MROWS, 256, 0, stream>>>(z2, cidx, labels, scale2, shift2, xc, csum);
    make_x1<<<MROWS, 256, 0, stream>>>(csum, counts, labels, x1);

    // GEMM3: out[2M,1024] = cat(xc,x1 | x1,xc) @ Wfc^T + bfc  (cat never materialized)
    wmma_gemm<2, false><<<dim3(gx, 4), 128, 0, stream>>>(
        xc, x1, Wfc, bfc, nullptr, nullptr, out, nullptr, nullptr,
        2 * MROWS, CATK, OUTSQ);
}